// FrameAggregator_62921270886589
// MI455X (gfx1250) — compile-verified
//
#include <hip/hip_runtime.h>
#include <hip/hip_bf16.h>
#include <math.h>

typedef __attribute__((ext_vector_type(2))) float v2f;
typedef __attribute__((ext_vector_type(8))) float v8f;

#define B_G   16
#define NPG_  256
#define T_    8
#define F_    64
#define H_    64
#define C_    32
#define O_    64
#define N_    4096      /* B_G*NPG_ */
#define E_    65536

// ---------------------------------------------------------------------------
// Wave-level fp32 WMMA: acc(16x16) += A(16xK) * B(Kx16), arbitrary strides.
// CDNA5 V_WMMA_F32_16X16X4_F32 layouts (ISA 7.12.2):
//   A: lane l (half=l>>4, l16=l&15): VGPR v = A[M=l16][K = k + 2*half + v]
//   B: VGPR v = B[K = k + 2*half + v][N = l16]
//   C/D: VGPR v = C[M = v + 8*half][N = l16]
// ---------------------------------------------------------------------------
__device__ __forceinline__ v8f wmma_acc(const float* __restrict__ A, long rsA, long csA,
                                        const float* __restrict__ B, long rsB, long csB,
                                        int K, v8f acc)
{
  const int lane = threadIdx.x & 31;
  const int half = lane >> 4;
  const int l16  = lane & 15;
  for (int k = 0; k < K; k += 4) {
    const int ka = k + 2 * half;
    v2f a, b;
    a[0] = A[(long)l16 * rsA + (long)(ka + 0) * csA];
    a[1] = A[(long)l16 * rsA + (long)(ka + 1) * csA];
    b[0] = B[(long)(ka + 0) * rsB + (long)l16 * csB];
    b[1] = B[(long)(ka + 1) * rsB + (long)l16 * csB];
    acc = __builtin_amdgcn_wmma_f32_16x16x4_f32(false, a, false, b,
                                                (short)0, acc, false, false);
  }
  return acc;
}

// ---------------------------------------------------------------------------
// Generic 2-level-batched strided GEMM (+ optional fused second GEMM,
// per-column bias, ReLU).  One wave per 16x16 output tile.
// C[nb,nt][m,n] = sum_k A[m,k]*B[k,n] (+ sum_k2 A2[m,k2]*B2[k2,n]) + bias[n]
// ---------------------------------------------------------------------------
__global__ void k_bgemm(const float* __restrict__ A,  long abS,  long atS,  long rsA,  long csA,
                        const float* __restrict__ Bm, long bbS,  long btS,  long rsB,  long csB,
                        const float* __restrict__ A2, long a2bS, long a2tS, long rsA2, long csA2,
                        const float* __restrict__ B2, long b2bS, long b2tS, long rsB2, long csB2,
                        float* __restrict__ C, long cbS, long ctS, long ldc,
                        const float* __restrict__ bias, int relu,
                        int M, int Ncol, int K, int K2, int NB, int NT)
{
  const int wid   = blockIdx.x * (blockDim.x >> 5) + (threadIdx.x >> 5);
  const int tn    = Ncol >> 4;
  const int tiles = (M >> 4) * tn;
  const int total = NB * NT * tiles;
  if (wid >= total) return;                 // wave-uniform exit (EXEC all-1s for WMMA)

  int bt   = wid / tiles;
  int tile = wid - bt * tiles;
  int nb   = bt / NT;
  int nt   = bt - nb * NT;
  int mt   = tile / tn;
  int ntc  = tile - mt * tn;
  long m0  = (long)mt << 4;
  long n0  = (long)ntc << 4;

  v8f acc = {0.f, 0.f, 0.f, 0.f, 0.f, 0.f, 0.f, 0.f};
  acc = wmma_acc(A  + nb * abS  + nt * atS  + m0 * rsA,  rsA,  csA,
                 Bm + nb * bbS  + nt * btS  + n0 * csB,  rsB,  csB, K, acc);
  if (A2)
    acc = wmma_acc(A2 + nb * a2bS + nt * a2tS + m0 * rsA2, rsA2, csA2,
                   B2 + nb * b2bS + nt * b2tS + n0 * csB2, rsB2, csB2, K2, acc);

  const int lane = threadIdx.x & 31;
  const int half = lane >> 4;
  const int l16  = lane & 15;
  const float bval = bias ? bias[n0 + l16] : 0.0f;
  float* Cp = C + nb * cbS + nt * ctS + m0 * ldc + n0;
#pragma unroll
  for (int v = 0; v < 8; ++v) {
    float val = acc[v] + bval;
    if (relu) val = fmaxf(val, 0.0f);
    Cp[(long)(v + 8 * half) * ldc + l16] = val;
  }
}

// --------------------------- small helper kernels --------------------------

__global__ void k_fill(float* p, float v, int n) {
  int i = blockIdx.x * blockDim.x + threadIdx.x;
  if (i < n) p[i] = v;
}

__global__ void k_deg_accum(const int* __restrict__ col, const float* __restrict__ ew,
                            float* __restrict__ deg) {
  int e = blockIdx.x * blockDim.x + threadIdx.x;
  if (e < E_) atomicAdd(deg + col[e], ew[e]);
}

__global__ void k_rsqrt_ip(float* p, int n) {
  int i = blockIdx.x * blockDim.x + threadIdx.x;
  if (i < n) { float d = p[i]; p[i] = d > 0.0f ? rsqrtf(d) : 0.0f; }
}

// One block per (edge | self-loop): h[dst] += xw[src] * norm over T_*H_ floats.
__global__ void k_scatter(const int* __restrict__ row, const int* __restrict__ col,
                          const float* __restrict__ ew, const float* __restrict__ dinv,
                          const float* __restrict__ xw, float* __restrict__ h)
{
  int e = blockIdx.x;
  int r, c; float norm;
  if (e < E_) {
    r = row[e]; c = col[e];
    norm = dinv[r] * ew[e] * dinv[c];
  } else {
    int i = e - E_;
    r = i; c = i;
    float d = dinv[i];
    norm = d * d;
  }
  const float* src = xw + (long)r * (T_ * H_);
  float*       dst = h  + (long)c * (T_ * H_);
  for (int j = threadIdx.x; j < T_ * H_; j += blockDim.x)
    atomicAdd(dst + j, src[j] * norm);
}

__global__ void k_adj_accum(const int* __restrict__ row, const int* __restrict__ col,
                            float* __restrict__ adj) {
  int e = blockIdx.x * blockDim.x + threadIdx.x;
  if (e < E_) {
    int rs = row[e], cs = col[e];
    int b = rs >> 8;  // NPG_ = 256
    atomicAdd(adj + ((long)b << 16) + ((long)(rs & 255) << 8) + (cs & 255), 1.0f);
  }
}

__global__ void k_bias_relu(float* __restrict__ h, const float* __restrict__ b1, int n) {
  int i = blockIdx.x * blockDim.x + threadIdx.x;
  if (i < n) h[i] = fmaxf(h[i] + b1[i & (H_ - 1)], 0.0f);
}

// dst[(b*NK+k)*NI+i] = mean_t src[b*bS + k*kS + t*tS + i]
__global__ void k_mean_t(const float* __restrict__ src, float* __restrict__ dst,
                         int NBo, int NK, int NI, long bS, long kS, long tS) {
  int i = blockIdx.x * blockDim.x + threadIdx.x;
  int tot = NBo * NK * NI;
  if (i >= tot) return;
  int ii = i % NI;
  int rest = i / NI;
  int k = rest % NK;
  int b = rest / NK;
  const float* p = src + (long)b * bS + (long)k * kS + ii;
  float s = 0.0f;
  for (int t = 0; t < T_; ++t) s += p[(long)t * tS];
  dst[i] = s * (1.0f / T_);
}

// in-place: row -> softmax(tanh(row))
__global__ void k_tanh_softmax(float* __restrict__ buf, int rows, int cols) {
  int r = blockIdx.x * blockDim.x + threadIdx.x;
  if (r >= rows) return;
  float* p = buf + (long)r * cols;
  float mx = -1e30f;
  for (int c = 0; c < cols; ++c) { float v = tanhf(p[c]); p[c] = v; mx = fmaxf(mx, v); }
  float s = 0.0f;
  for (int c = 0; c < cols; ++c) { float e = expf(p[c] - mx); p[c] = e; s += e; }
  float inv = 1.0f / s;
  for (int c = 0; c < cols; ++c) p[c] *= inv;
}

__global__ void k_rowsum(const float* __restrict__ src, float* __restrict__ dst,
                         int rows, int cols) {
  int r = blockIdx.x * blockDim.x + threadIdx.x;
  if (r >= rows) return;
  const float* p = src + (long)r * cols;
  float s = 0.0f;
  for (int c = 0; c < cols; ++c) s += p[c];
  dst[r] = s;
}

__device__ __forceinline__ float block_reduce_256(float v, float* red) {
  int tid = threadIdx.x;
  red[tid] = v; __syncthreads();
  for (int st = 128; st > 0; st >>= 1) {
    if (tid < st) red[tid] += red[tid + st];
    __syncthreads();
  }
  float r = red[0];
  __syncthreads();
  return r;
}

// Per graph b: mincut num=trace(out_adj), den=sum_n d[n]*sum_k s[n,k]^2,
// ortho = || ss/||ss||_F - I/sqrt(k) ||_F
__global__ void k_pool_scalars(const float* __restrict__ oadj, const float* __restrict__ dflat,
                               const float* __restrict__ s, const float* __restrict__ ss,
                               int n, int k,
                               float* __restrict__ num, float* __restrict__ den,
                               float* __restrict__ orth)
{
  __shared__ float red[256];
  __shared__ float froS;
  int b = blockIdx.x, tid = threadIdx.x;
  const float* oa  = oadj + (long)b * k * k;
  const float* sb  = s    + (long)b * n * k;
  const float* ssb = ss   + (long)b * k * k;

  float loc = 0.0f;
  for (int i = tid; i < k; i += 256) loc += oa[(long)i * k + i];
  float r = block_reduce_256(loc, red);
  if (tid == 0) num[b] = r;

  loc = 0.0f;
  for (int i = tid; i < n; i += 256) {
    const float* sr = sb + (long)i * k;
    float a = 0.0f;
    for (int c = 0; c < k; ++c) a += sr[c] * sr[c];
    loc += dflat[(long)b * n + i] * a;
  }
  r = block_reduce_256(loc, red);
  if (tid == 0) den[b] = r;

  int kk = k * k;
  loc = 0.0f;
  for (int i = tid; i < kk; i += 256) { float v = ssb[i]; loc += v * v; }
  r = block_reduce_256(loc, red);
  if (tid == 0) froS = sqrtf(r);
  __syncthreads();
  float fro = froS;
  float tgt = rsqrtf((float)k);
  loc = 0.0f;
  for (int i = tid; i < kk; i += 256) {
    int ri = i / k, ci = i - ri * k;
    float v = ssb[i] / fro - ((ri == ci) ? tgt : 0.0f);
    loc += v * v;
  }
  r = block_reduce_256(loc, red);
  if (tid == 0) orth[b] = sqrtf(r);
}

// zero diag; d=sqrt(rowsum)+EPS; oadj[i][j] /= d[i]*d[j].  blockDim == k (<=64).
__global__ void k_norm_adj(float* __restrict__ oadj, int k) {
  __shared__ float dsh[64];
  int b = blockIdx.x, i = threadIdx.x;
  float* oa = oadj + (long)b * k * k;
  oa[(long)i * k + i] = 0.0f;
  __syncthreads();
  float s = 0.0f;
  for (int j = 0; j < k; ++j) s += oa[(long)i * k + j];
  dsh[i] = sqrtf(s) + 1e-15f;
  __syncthreads();
  float di = dsh[i];
  for (int j = 0; j < k; ++j) oa[(long)i * k + j] /= (di * dsh[j]);
}

// sc: num1[0..15] den1[16..31] orth1[32..47] num2[48..63] den2[64..79] orth2[80..95]
__global__ void k_losses(const float* __restrict__ sc, float* __restrict__ out) {
  if (threadIdx.x == 0 && blockIdx.x == 0) {
    float mc = 0.0f, oo = 0.0f;
    for (int b = 0; b < B_G; ++b) {
      mc += -(sc[b] / sc[16 + b]) - (sc[48 + b] / sc[64 + b]);
      oo += sc[32 + b] + sc[80 + b];
    }
    out[0] = mc / (float)B_G;
    out[1] = oo / (float)B_G;
  }
}

// ------------------------------- host side ---------------------------------

static inline void bgemm(hipStream_t st,
                         const float* A, long abS, long atS, long rsA, long csA,
                         const float* Bm, long bbS, long btS, long rsB, long csB,
                         const float* A2, long a2bS, long a2tS, long rsA2, long csA2,
                         const float* B2, long b2bS, long b2tS, long rsB2, long csB2,
                         float* C, long cbS, long ctS, long ldc,
                         const float* bias, int relu,
                         int M, int Ncol, int K, int K2, int NB, int NT)
{
  int waves = NB * NT * (M / 16) * (Ncol / 16);
  int blocks = (waves + 7) / 8;
  k_bgemm<<<blocks, 256, 0, st>>>(A, abS, atS, rsA, csA, Bm, bbS, btS, rsB, csB,
                                  A2, a2bS, a2tS, rsA2, csA2, B2, b2bS, b2tS, rsB2, csB2,
                                  C, cbS, ctS, ldc, bias, relu, M, Ncol, K, K2, NB, NT);
}

extern "C" void kernel_launch(void* const* d_in, const int* in_sizes, int n_in,
                              void* d_out, int out_size, void* d_ws, size_t ws_size,
                              hipStream_t stream)
{
  (void)in_sizes; (void)n_in; (void)out_size; (void)ws_size;
  const float* pos   = (const float*)d_in[0];
  const int*   eidx  = (const int*)  d_in[1];
  const float* eattr = (const float*)d_in[2];
  /* d_in[3] = batch: implied by node index (batch[i] = i/256) */
  const float* W1   = (const float*)d_in[4];
  const float* b1   = (const float*)d_in[5];
  const float* p1W  = (const float*)d_in[6];
  const float* p1b  = (const float*)d_in[7];
  const float* c2rW = (const float*)d_in[8];
  const float* c2rb = (const float*)d_in[9];
  const float* c2sW = (const float*)d_in[10];
  const float* p2W  = (const float*)d_in[11];
  const float* p2b  = (const float*)d_in[12];
  const float* c3rW = (const float*)d_in[13];
  const float* c3rb = (const float*)d_in[14];
  const float* c3sW = (const float*)d_in[15];
  const int* row = eidx;
  const int* col = eidx + E_;
  float* out = (float*)d_out;

  // ---- workspace layout (floats) ----
  float* ws     = (float*)d_ws;
  float* deg    = ws + 0;        // 4096 (becomes dinv)
  float* dflat1 = ws + 4096;     // 4096
  float* dflat2 = ws + 8192;     // 1024
  float* sc     = ws + 9216;     // 96 scalar slots
  float* xw     = ws + 16384;    // 2,097,152 (reused region after scatter)
  float* h      = ws + 2113536;  // 2,097,152  [N,T,H]
  float* adj    = ws + 4210688;  // 1,048,576  [B,256,256]
  float* s1     = ws + 5259264;  // 262,144    [B,256,64]
  float* x1     = ws + 5521408;  // 524,288    [B,T,64,64]
  float* AS1    = ws + 6045696;  // 262,144    [B,256,64] (later: ss buffers)
  float* oadj1  = ws + 6307840;  // 65,536     [B,64,64]
  // reuse of the xw region once the edge scatter has consumed it:
  float* xm    = xw;              // 262,144 (xm1 [4096,64]; later xm2 [1024,64])
  float* tmp1  = xw + 262144;     // 524,288 [B,T,64,64]
  float* x2    = xw + 786432;     // 524,288 [B,T,64,64]
  float* x3    = xw + 1310720;    // 262,144 [B,T,32,64]
  float* tmp2  = xw + 1572864;    // 262,144 [B,T,32,64]
  float* AS2   = xw + 1835008;    // 32,768  [B,64,32]
  float* s2    = xw + 1867776;    // 32,768  [B,64,32]
  float* oadj2 = xw + 1900544;    // 16,384  [B,32,32]
  float* ss1   = AS1;             // 65,536 (AS1 free after oadj1)
  float* ss2   = AS1 + 65536;     // 16,384

  // ---- 0) zero accumulation targets ----
  hipMemsetAsync(h,   0, (size_t)N_ * T_ * H_ * sizeof(float), stream);
  hipMemsetAsync(adj, 0, (size_t)B_G * NPG_ * NPG_ * sizeof(float), stream);

  // ---- 1) degrees: deg = 1 (self loop) + sum_{e: dst==i} w[e]; dinv = rsqrt ----
  k_fill<<<16, 256, 0, stream>>>(deg, 1.0f, N_);
  k_deg_accum<<<E_ / 256, 256, 0, stream>>>(col, eattr, deg);
  k_rsqrt_ip<<<16, 256, 0, stream>>>(deg, N_);

  // ---- 2) xw = pos @ W1   [32768,64]x[64,64]  (WMMA) ----
  bgemm(stream, pos, 0, 0, F_, 1,  W1, 0, 0, H_, 1,
        nullptr, 0,0,0,0, nullptr, 0,0,0,0,
        xw, 0, 0, H_, nullptr, 0, N_ * T_, H_, F_, 0, 1, 1);

  // ---- 3) GCN scatter (edges + self loops) and dense adjacency ----
  k_scatter<<<E_ + N_, 256, 0, stream>>>(row, col, eattr, deg, xw, h);
  k_adj_accum<<<E_ / 256, 256, 0, stream>>>(row, col, adj);
  k_bias_relu<<<(N_ * T_ * H_) / 256, 256, 0, stream>>>(h, b1, N_ * T_ * H_);

  // ---- 4) s1 = softmax(tanh(mean_t(x) @ p1W + p1b)) ----
  k_mean_t<<<(N_ * H_) / 256, 256, 0, stream>>>(h, xm, N_, 1, H_, (long)T_ * H_, 0, H_);
  bgemm(stream, xm, 0, 0, H_, 1,  p1W, 0, 0, 2 * C_, 1,
        nullptr, 0,0,0,0, nullptr, 0,0,0,0,
        s1, 0, 0, 2 * C_, p1b, 0, N_, 2 * C_, H_, 0, 1, 1);
  k_tanh_softmax<<<N_ / 256, 256, 0, stream>>>(s1, N_, 2 * C_);

  // ---- 5) mincut pool 1 ----
  // x1[b,t] = S1^T(64x256) @ X(256x64)
  bgemm(stream, s1, (long)NPG_ * 64, 0, 1, 64,
        h, (long)NPG_ * T_ * H_, H_, (long)T_ * H_, 1,
        nullptr, 0,0,0,0, nullptr, 0,0,0,0,
        x1, (long)T_ * 64 * H_, 64 * H_, H_, nullptr, 0,
        64, H_, NPG_, 0, B_G, T_);
  // AS1 = adj @ S1 ; oadj1 = S1^T @ AS1 ; ss1 = S1^T @ S1
  bgemm(stream, adj, (long)NPG_ * NPG_, 0, NPG_, 1,  s1, (long)NPG_ * 64, 0, 64, 1,
        nullptr, 0,0,0,0, nullptr, 0,0,0,0,
        AS1, (long)NPG_ * 64, 0, 64, nullptr, 0, NPG_, 64, NPG_, 0, B_G, 1);
  bgemm(stream, s1, (long)NPG_ * 64, 0, 1, 64,  AS1, (long)NPG_ * 64, 0, 64, 1,
        nullptr, 0,0,0,0, nullptr, 0,0,0,0,
        oadj1, 64 * 64, 0, 64, nullptr, 0, 64, 64, NPG_, 0, B_G, 1);
  bgemm(stream, s1, (long)NPG_ * 64, 0, 1, 64,  s1, (long)NPG_ * 64, 0, 64, 1,
        nullptr, 0,0,0,0, nullptr, 0,0,0,0,
        ss1, 64 * 64, 0, 64, nullptr, 0, 64, 64, NPG_, 0, B_G, 1);
  k_rowsum<<<N_ / 256, 256, 0, stream>>>(adj, dflat1, B_G * NPG_, NPG_);
  k_pool_scalars<<<B_G, 256, 0, stream>>>(oadj1, dflat1, s1, ss1, NPG_, 64,
                                          sc + 0, sc + 16, sc + 32);
  k_norm_adj<<<B_G, 64, 0, stream>>>(oadj1, 64);

  // ---- 6) dense graph conv 2: x2 = relu((adj1n@x1)@c2rW + x1@c2sW + c2rb) ----
  bgemm(stream, oadj1, 64 * 64, 0, 64, 1,
        x1, (long)T_ * 64 * H_, 64 * H_, H_, 1,
        nullptr, 0,0,0,0, nullptr, 0,0,0,0,
        tmp1, (long)T_ * 64 * H_, 64 * H_, H_, nullptr, 0, 64, H_, 64, 0, B_G, T_);
  bgemm(stream, tmp1, (long)T_ * 64 * H_, 64 * H_, H_, 1,  c2rW, 0, 0, H_, 1,
        x1,   (long)T_ * 64 * H_, 64 * H_, H_, 1,          c2sW, 0, 0, H_, 1,
        x2, (long)T_ * 64 * H_, 64 * H_, H_, c2rb, 1, 64, H_, H_, H_, B_G, T_);

  // ---- 7) s2 = softmax(tanh(mean_t(x2) @ p2W + p2b)) ----
  k_mean_t<<<(B_G * 64 * H_) / 256, 256, 0, stream>>>(x2, xm, B_G, 64, H_,
                                                      (long)T_ * 64 * H_, H_, 64 * H_);
  bgemm(stream, xm, 0, 0, H_, 1,  p2W, 0, 0, C_, 1,
        nullptr, 0,0,0,0, nullptr, 0,0,0,0,
        s2, 0, 0, C_, p2b, 0, B_G * 64, C_, H_, 0, 1, 1);
  k_tanh_softmax<<<(B_G * 64) / 256, 256, 0, stream>>>(s2, B_G * 64, C_);

  // ---- 8) mincut pool 2 ----
  bgemm(stream, s2, 64 * C_, 0, 1, C_,
        x2, (long)T_ * 64 * H_, 64 * H_, H_, 1,
        nullptr, 0,0,0,0, nullptr, 0,0,0,0,
        x3, (long)T_ * C_ * H_, C_ * H_, H_, nullptr, 0, C_, H_, 64, 0, B_G, T_);
  bgemm(stream, oadj1, 64 * 64, 0, 64, 1,  s2, 64 * C_, 0, C_, 1,
        nullptr, 0,0,0,0, nullptr, 0,0,0,0,
        AS2, 64 * C_, 0, C_, nullptr, 0, 64, C_, 64, 0, B_G, 1);
  bgemm(stream, s2, 64 * C_, 0, 1, C_,  AS2, 64 * C_, 0, C_, 1,
        nullptr, 0,0,0,0, nullptr, 0,0,0,0,
        oadj2, C_ * C_, 0, C_, nullptr, 0, C_, C_, 64, 0, B_G, 1);
  bgemm(stream, s2, 64 * C_, 0, 1, C_,  s2, 64 * C_, 0, C_, 1,
        nullptr, 0,0,0,0, nullptr, 0,0,0,0,
        ss2, C_ * C_, 0, C_, nullptr, 0, C_, C_, 64, 0, B_G, 1);
  k_rowsum<<<(B_G * 64) / 256, 256, 0, stream>>>(oadj1, dflat2, B_G * 64, 64);
  k_pool_scalars<<<B_G, 256, 0, stream>>>(oadj2, dflat2, s2, ss2, 64, C_,
                                          sc + 48, sc + 64, sc + 80);
  k_norm_adj<<<B_G, C_, 0, stream>>>(oadj2, C_);

  // ---- 9) final conv: out_x = (adj2n@x3)@c3rW + x3@c3sW + c3rb (no relu) ----
  bgemm(stream, oadj2, C_ * C_, 0, C_, 1,
        x3, (long)T_ * C_ * H_, C_ * H_, H_, 1,
        nullptr, 0,0,0,0, nullptr, 0,0,0,0,
        tmp2, (long)T_ * C_ * H_, C_ * H_, H_, nullptr, 0, C_, H_, C_, 0, B_G, T_);
  bgemm(stream, tmp2, (long)T_ * C_ * H_, C_ * H_, H_, 1,  c3rW, 0, 0, O_, 1,
        x3,   (long)T_ * C_ * H_, C_ * H_, H_, 1,          c3sW, 0, 0, O_, 1,
        out, (long)T_ * C_ * O_, C_ * O_, O_, c3rb, 0, C_, O_, H_, H_, B_G, T_);

  // ---- 10) losses and agg = softmax(s1) @ softmax(s2) ----
  k_losses<<<1, 32, 0, stream>>>(sc, out + B_G * T_ * C_ * O_);
  bgemm(stream, s1, (long)NPG_ * 64, 0, 64, 1,  s2, 64 * C_, 0, C_, 1,
        nullptr, 0,0,0,0, nullptr, 0,0,0,0,
        out + B_G * T_ * C_ * O_ + 2, (long)NPG_ * C_, 0, C_,
        nullptr, 0, NPG_, C_, 64, 0, B_G, 1);
}